// SNNLayer_65790309040242
// MI455X (gfx1250) — compile-verified
//
#include <hip/hip_runtime.h>
#include <hip/hip_bf16.h>

#define NB 32
#define NN 1024
#define NO 1024
#define BIGV 1e10f

typedef __attribute__((ext_vector_type(2))) float v2f;
typedef __attribute__((ext_vector_type(8))) float v8f;

// ---------------------------------------------------------------------------
// Kernel 1: per-batch bitonic argsort (ascending) in LDS.
// One block of 1024 threads per batch row.
// ---------------------------------------------------------------------------
__global__ __launch_bounds__(1024) void snn_sort_kernel(const float* __restrict__ in,
                                                        float* __restrict__ xs,
                                                        int* __restrict__ ord) {
    __shared__ float key[NN];
    __shared__ int   val[NN];
    const int b = blockIdx.x;
    const int t = threadIdx.x;
    key[t] = in[b * NN + t];
    val[t] = t;
    __syncthreads();
    for (int k = 2; k <= NN; k <<= 1) {
        for (int j = k >> 1; j > 0; j >>= 1) {
            int ixj = t ^ j;
            if (ixj > t) {
                bool up = ((t & k) == 0);
                float ka = key[t], kb = key[ixj];
                bool sw = up ? (ka > kb) : (ka < kb);
                if (sw) {
                    int va = val[t];
                    key[t] = kb; val[t] = val[ixj];
                    key[ixj] = ka; val[ixj] = va;
                }
            }
            __syncthreads();
        }
    }
    xs[b * NN + t]  = key[t];
    ord[b * NN + t] = val[t];
}

// ---------------------------------------------------------------------------
// Kernel 2: row maxima of w (prefilter). If 2*max(w[o,:]) <= 1, no adjacent
// pair sum can exceed 1.0 for any batch => out[b,o] == BIG without scanning.
// One 256-thread block per row.
// ---------------------------------------------------------------------------
__global__ __launch_bounds__(256) void snn_rowmax_kernel(const float* __restrict__ w,
                                                         float* __restrict__ rmax) {
    const int o = blockIdx.x;
    const int t = threadIdx.x;
    float m = -3.4e38f;
    for (int i = t; i < NN; i += 256) m = fmaxf(m, w[(size_t)o * NN + i]);
    for (int off = 16; off > 0; off >>= 1) m = fmaxf(m, __shfl_down(m, off, 32));
    __shared__ float sm[8];
    if ((t & 31) == 0) sm[t >> 5] = m;
    __syncthreads();
    if (t == 0) {
        float mm = sm[0];
        for (int i = 1; i < 8; ++i) mm = fmaxf(mm, sm[i]);
        rmax[o] = mm;
    }
}

// ---------------------------------------------------------------------------
// Kernel 3: WMMA-tiled first-hit scan.
// Each wave owns 16 outputs (M dim). For each 16-position column tile, the
// adjacent-pair sums are computed as Wg(16x20) x Bband(20x16) via chained
// V_WMMA_F32_16X16X4_F32 (5 K-chunks; rows j>16 and the p=-1 row zeroed).
//   w_sum  = Wg x Bw,  Bw[j,i] = [j==i] + [j==i+1]
//   mul_sum= Wg x Bm,  Bm[j,i] = x[p(j)] * Bw[j,i]     (p(j) = tile*16 + j - 1)
// All out-of-range contributions are masked through Bw (no divergent loads:
// lx/lo are indexed with the clamped position, and Bw==0 wherever p is
// invalid, so the masked value never reaches the accumulator).
// First qualifying column per output found with wave32 ballot/ctz/shfl using
// the documented C/D layout (lane = column; VGPR r = rows r and r+8).
// ---------------------------------------------------------------------------
__global__ __launch_bounds__(256) void snn_scan_kernel(const float* __restrict__ w,
                                                       const float* __restrict__ xs,
                                                       const int* __restrict__ ord,
                                                       const float* __restrict__ rmax,
                                                       float* __restrict__ out) {
    const int b     = blockIdx.y;
    const int wave  = threadIdx.x >> 5;
    const int lane  = threadIdx.x & 31;
    const int obase = blockIdx.x * 128 + wave * 16;

    __shared__ float lx[NN];
    __shared__ int   lo[NN];
    for (int i = threadIdx.x; i < NN; i += 256) {
        lx[i] = xs[(size_t)b * NN + i];
        lo[i] = ord[(size_t)b * NN + i];
    }
    __syncthreads();

    const int m16   = lane & 15;   // A-row m / B-column n / D-column n
    const int khalf = lane >> 4;   // lane half selects K pair

    // Prefilter: can any of this wave's 16 outputs ever satisfy w_sum > 1 ?
    bool canlane = (lane < 16) ? (2.0f * rmax[obase + lane] > 1.0f) : false;
    unsigned can = __builtin_amdgcn_ballot_w32(canlane);

    bool  myDone = (lane >= 16);   // lanes 0..15 own outputs obase+lane
    float myVal  = 0.0f;

    if (can != 0u) {
        const float* wrow = w + (size_t)(obase + m16) * NN;
        __builtin_prefetch(wrow, 0, 1);   // global_prefetch_b8

        // Hoisted band pattern: itile-invariant part of Bw (row j hits
        // columns j-1 and j; rows j>16 are K-padding and always zero).
        float bwBase[5][2];
        int   jIdx[5][2];
#pragma unroll
        for (int chunk = 0; chunk < 5; ++chunk) {
#pragma unroll
            for (int v = 0; v < 2; ++v) {
                const int j = chunk * 4 + 2 * khalf + v;
                jIdx[chunk][v] = j;
                bwBase[chunk][v] =
                    (((j == m16) || (j == m16 + 1)) && (j <= 16)) ? 1.0f : 0.0f;
            }
        }

        for (int itile = 0; itile < NN / 16; ++itile) {
            v8f accW = {};   // w_sum tile
            v8f accM = {};   // mul_sum tile
#pragma unroll
            for (int chunk = 0; chunk < 5; ++chunk) {
                v2f a, bw, bm;
#pragma unroll
                for (int v = 0; v < 2; ++v) {
                    const int j = jIdx[chunk][v];
                    const int p = itile * 16 + j - 1;    // global sorted pos
                    int pa = p;
                    if (pa < 0) pa = 0;
                    if (pa > NN - 1) pa = NN - 1;
                    const float av  = wrow[lo[pa]];      // Wg gather (A)
                    // drop the p==-1 row (first global position has no i-1)
                    const float bwv = (p >= 0) ? bwBase[chunk][v] : 0.0f;
                    const float xv  = lx[pa];            // masked by bwv
                    a[v]  = av;
                    bw[v] = bwv;
                    bm[v] = bwv * xv;
                }
                accW = __builtin_amdgcn_wmma_f32_16x16x4_f32(
                    false, a, false, bw, (short)0, accW, false, false);
                accM = __builtin_amdgcn_wmma_f32_16x16x4_f32(
                    false, a, false, bm, (short)0, accM, false, false);
            }

            const float xi = lx[itile * 16 + m16];   // x at this lane's column
#pragma unroll
            for (int r = 0; r < 8; ++r) {
                const float ws  = accW[r];
                const float ms  = accM[r];
                const float den = fminf(fmaxf(ws - 1.0f, 1e-10f), BIGV);
                const float oa  = ms / den;
                const bool cond = (oa > xi) && (ws > 1.0f);
                const unsigned mask = __builtin_amdgcn_ballot_w32(cond);
                const unsigned lowm = mask & 0xFFFFu;   // rows m = r
                const unsigned him  = mask >> 16;       // rows m = r + 8
                if (lowm) {
                    const int il = __builtin_ctz(lowm);
                    const float v0 = __shfl(oa, il, 32);
                    if (lane == r && !myDone) { myDone = true; myVal = v0; }
                }
                if (him) {
                    const int il = __builtin_ctz(him);
                    const float v1 = __shfl(oa, 16 + il, 32);
                    if (lane == r + 8 && !myDone) { myDone = true; myVal = v1; }
                }
            }
            if (__builtin_amdgcn_ballot_w32(myDone) == 0xFFFFFFFFu) break;
        }
    }

    if (lane < 16) {
        out[(size_t)b * NO + obase + lane] = myDone ? myVal : BIGV;
    }
}

// ---------------------------------------------------------------------------
extern "C" void kernel_launch(void* const* d_in, const int* in_sizes, int n_in,
                              void* d_out, int out_size, void* d_ws, size_t ws_size,
                              hipStream_t stream) {
    (void)in_sizes; (void)n_in; (void)out_size; (void)ws_size;
    const float* inp = (const float*)d_in[0];   // [B,N] f32
    const float* w   = (const float*)d_in[1];   // [O,N] f32
    float* out = (float*)d_out;                 // [B,O] f32

    float* xs   = (float*)d_ws;                                       // B*N f32
    int*   ordp = (int*)((char*)d_ws + (size_t)NB * NN * sizeof(float));
    float* rmax = (float*)((char*)d_ws + 2ull * NB * NN * sizeof(float)); // O f32

    snn_sort_kernel<<<NB, NN, 0, stream>>>(inp, xs, ordp);
    snn_rowmax_kernel<<<NO, 256, 0, stream>>>(w, rmax);
    snn_scan_kernel<<<dim3(NO / 128, NB), 256, 0, stream>>>(w, xs, ordp, rmax, out);
}